// OrientationLoss_26456998543433
// MI455X (gfx1250) — compile-verified
//
#include <hip/hip_runtime.h>
#include <stdint.h>

// ---------------------------------------------------------------------------
// OrientationLoss on MI455X (gfx1250): memory-bound streaming reduction.
// 600 B/sample, 1M samples -> ~600 MB @ 23.3 TB/s ~ 26 us floor.
// Tiles of 128 samples are staged global->LDS with CDNA5 async-to-LDS b128
// loads (double buffered), compute reads LDS, two-level deterministic
// reduction (wave32 shuffle -> block partial doubles -> finalize kernel).
// ---------------------------------------------------------------------------

#define BLOCK 128
#define TILE 128
#define PRED_F 72                                  // floats per sample (24*3)
#define TARG_F 78                                  // floats per sample (26*3)
#define PRED_TILE_F (TILE * PRED_F)                // 9216 floats
#define TARG_TILE_F (TILE * TARG_F)                // 9984 floats
#define PRED_CHUNKS (PRED_TILE_F / 4)              // 2304 float4
#define TARG_CHUNKS (TARG_TILE_F / 4)              // 2496 float4
#define PRED_ITER (PRED_CHUNKS / BLOCK)            // 18 per thread
#define TARG_ITER ((TARG_CHUNKS + BLOCK - 1) / BLOCK) // 20 per thread (clamped)
#define ASYNC_N (PRED_ITER + TARG_ITER)            // 38 async ops per thread
#define RED_OFF (2 * PRED_TILE_F + 2 * TARG_TILE_F)
#define LDS_FLOATS (RED_OFF + 16)                  // + reduction scratch

#define AS1 __attribute__((address_space(1)))
#define AS3 __attribute__((address_space(3)))

typedef int v4i __attribute__((ext_vector_type(4)));

#if defined(__HIP_DEVICE_COMPILE__) && __has_builtin(__builtin_amdgcn_global_load_async_to_lds_b128)
#define HAS_ASYNC_LDS 1
#else
#define HAS_ASYNC_LDS 0
#endif

#if defined(__HIP_DEVICE_COMPILE__) && __has_builtin(__builtin_amdgcn_s_wait_asynccnt)
#define WAIT_ASYNC(n) __builtin_amdgcn_s_wait_asynccnt(n)
#elif HAS_ASYNC_LDS
#define WAIT_ASYNC(n) asm volatile("s_wait_asynccnt %0" ::"i"(n) : "memory")
#else
#define WAIT_ASYNC(n) do {} while (0)
#endif

// bone-direction joint tables (compile-time constants from the reference)
static constexpr int kPA[13] = {2, 14, 6, 5, 4, 8, 7, 11, 10, 19, 18, 21, 20};
static constexpr int kPB[13] = {1, 13, 3, 2, 1, 5, 4, 8, 7, 17, 16, 19, 18};
static constexpr int kTA[13] = {22, 11, 2, 23, 19, 24, 20, 25, 21, 13, 6, 14, 7};
static constexpr int kTB[13] = {18, 4, 1, 22, 18, 23, 19, 24, 20, 12, 5, 13, 6};

__device__ __forceinline__ void sub3(const float* b, int ja, int jb, float o[3]) {
  o[0] = b[3 * ja + 0] - b[3 * jb + 0];
  o[1] = b[3 * ja + 1] - b[3 * jb + 1];
  o[2] = b[3 * ja + 2] - b[3 * jb + 2];
}

__device__ __forceinline__ void cross3(const float a[3], const float b[3], float o[3]) {
  o[0] = a[1] * b[2] - a[2] * b[1];
  o[1] = a[2] * b[0] - a[0] * b[2];
  o[2] = a[0] * b[1] - a[1] * b[0];
}

__device__ __forceinline__ float cos3(const float a[3], const float b[3]) {
  float num = a[0] * b[0] + a[1] * b[1] + a[2] * b[2];
  float na = a[0] * a[0] + a[1] * a[1] + a[2] * a[2];
  float nb = b[0] * b[0] + b[1] * b[1] + b[2] * b[2];
  return num / sqrtf(na * nb);
}

__device__ __forceinline__ void computeSample(const float* __restrict__ P,
                                              const float* __restrict__ T,
                                              float sc, const int* si, const int* di,
                                              float& accL1, float& accCos) {
  float l1 = 0.f;
#pragma unroll
  for (int k = 0; k < 14; ++k) {
    const int pj = si[k] * 3, tj = di[k] * 3;
#pragma unroll
    for (int c = 0; c < 3; ++c) {
      float d = sc * P[pj + c] - T[tj + c];
      float ad = fabsf(d);
      l1 += (ad < 1.f) ? 0.5f * d * d : (ad - 0.5f);
    }
  }
  float cs = 0.f;
  {
    float e1[3], e2[3], f1[3], f2[3], pc[3], tc[3];
    // pelvis: cross(P2-P0, P1-P0) vs cross(T22-T1, T18-T1)
    sub3(P, 2, 0, e1); sub3(P, 1, 0, e2); cross3(e1, e2, pc);
    sub3(T, 22, 1, f1); sub3(T, 18, 1, f2); cross3(f1, f2, tc);
    cs += cos3(pc, tc);
    // chest: cross(P13-P9, P14-P9) vs cross(T3-T2, T11-T2)
    sub3(P, 13, 9, e1); sub3(P, 14, 9, e2); cross3(e1, e2, pc);
    sub3(T, 3, 2, f1); sub3(T, 11, 2, f2); cross3(f1, f2, tc);
    cs += cos3(pc, tc);
  }
#pragma unroll
  for (int k = 0; k < 13; ++k) {
    float vp[3], vt[3];
    sub3(P, kPA[k], kPB[k], vp);
    sub3(T, kTA[k], kTB[k], vt);
    cs += cos3(vp, vt);
  }
  accL1 += l1;
  accCos += cs;
}

__device__ __forceinline__ float waveReduce(float v) {
#pragma unroll
  for (int m = 16; m; m >>= 1) v += __shfl_xor(v, m, 32);
  return v;
}

__global__ void __launch_bounds__(BLOCK)
orientation_loss_main(const float* __restrict__ scale,
                      const int* __restrict__ si_g,
                      const int* __restrict__ di_g,
                      const float* __restrict__ pred,
                      const float* __restrict__ targ,
                      double* __restrict__ partial,
                      long long B) {
  extern __shared__ float lds[];
  float* ldsP = lds;                       // [2][PRED_TILE_F]
  float* ldsT = lds + 2 * PRED_TILE_F;     // [2][TARG_TILE_F]
  float* red  = lds + RED_OFF;             // [8]

  const int tid = threadIdx.x;
  const int G = gridDim.x;
  const float sc = scale[0];
  int si[14], di[14];
#pragma unroll
  for (int k = 0; k < 14; ++k) { si[k] = si_g[k]; di[k] = di_g[k]; }

  const int nFull = (int)(B / TILE);
  float accL1 = 0.f, accCos = 0.f;

  // stage one 128-sample tile into LDS buffer `buf` (coalesced b128 stream)
  auto stage = [&](int buf, int t) {
    const float* pbase = pred + (size_t)t * PRED_TILE_F;
    const float* tbase = targ + (size_t)t * TARG_TILE_F;
    float* lp = ldsP + buf * PRED_TILE_F;
    float* lt = ldsT + buf * TARG_TILE_F;
#if HAS_ASYNC_LDS
#pragma unroll
    for (int k = 0; k < PRED_ITER; ++k) {
      const int c = tid + k * BLOCK;  // exact cover: 18*128 == 2304
      __builtin_amdgcn_global_load_async_to_lds_b128(
          (AS1 v4i*)(pbase + (size_t)c * 4), (AS3 v4i*)(lp + (size_t)c * 4), 0, 0);
    }
#pragma unroll
    for (int k = 0; k < TARG_ITER; ++k) {
      int c = tid + k * BLOCK;
      c = (c > TARG_CHUNKS - 1) ? (TARG_CHUNKS - 1) : c;  // uniform 20 issues/thread
      __builtin_amdgcn_global_load_async_to_lds_b128(
          (AS1 v4i*)(tbase + (size_t)c * 4), (AS3 v4i*)(lt + (size_t)c * 4), 0, 0);
    }
#else
#pragma unroll
    for (int k = 0; k < PRED_ITER; ++k) {
      const int c = tid + k * BLOCK;
      *(float4*)(lp + (size_t)c * 4) = *(const float4*)(pbase + (size_t)c * 4);
    }
#pragma unroll
    for (int k = 0; k < TARG_ITER; ++k) {
      int c = tid + k * BLOCK;
      c = (c > TARG_CHUNKS - 1) ? (TARG_CHUNKS - 1) : c;
      *(float4*)(lt + (size_t)c * 4) = *(const float4*)(tbase + (size_t)c * 4);
    }
#endif
  };

  // double-buffered pipeline over full tiles (grid-stride)
  if (blockIdx.x < (unsigned)nFull) stage(0, blockIdx.x);
  int i = 0;
  for (int cur = blockIdx.x; cur < nFull; cur += G, ++i) {
    const int buf = i & 1;
    const int nxt = cur + G;
    if (nxt < nFull) {
      stage(buf ^ 1, nxt);
      WAIT_ASYNC(ASYNC_N);   // previous batch (<=38) fully retired, in order
    } else {
      WAIT_ASYNC(0);
    }
    __syncthreads();         // all waves' stages for `buf` are visible
    const float* P = ldsP + buf * PRED_TILE_F + tid * PRED_F;
    const float* T = ldsT + buf * TARG_TILE_F + tid * TARG_F;
    computeSample(P, T, sc, si, di, accL1, accCos);
    __syncthreads();         // done reading `buf` before it is restaged
  }

  // tail samples (< TILE), straight from global, block 0 only
  const long long tailBase = (long long)nFull * TILE;
  const int tail = (int)(B - tailBase);
  if (blockIdx.x == 0 && tid < tail) {
    computeSample(pred + (size_t)(tailBase + tid) * PRED_F,
                  targ + (size_t)(tailBase + tid) * TARG_F,
                  sc, si, di, accL1, accCos);
  }

  // deterministic block reduction: wave32 shuffle tree, then 4 partials in LDS
  accL1 = waveReduce(accL1);
  accCos = waveReduce(accCos);
  const int wave = tid >> 5, lane = tid & 31;
  if (lane == 0) { red[wave * 2] = accL1; red[wave * 2 + 1] = accCos; }
  __syncthreads();
  if (tid == 0) {
    double s0 = 0.0, s1 = 0.0;
#pragma unroll
    for (int w = 0; w < BLOCK / 32; ++w) {
      s0 += (double)red[w * 2];
      s1 += (double)red[w * 2 + 1];
    }
    partial[(size_t)blockIdx.x * 2 + 0] = s0;
    partial[(size_t)blockIdx.x * 2 + 1] = s1;
  }
}

__global__ void __launch_bounds__(256)
orientation_loss_final(const double* __restrict__ partial, int nPart,
                       float* __restrict__ out, long long B) {
  __shared__ double sh[512];
  double s0 = 0.0, s1 = 0.0;
  for (int i = threadIdx.x; i < nPart; i += 256) {
    s0 += partial[(size_t)i * 2 + 0];
    s1 += partial[(size_t)i * 2 + 1];
  }
  sh[threadIdx.x] = s0;
  sh[256 + threadIdx.x] = s1;
  __syncthreads();
  for (int off = 128; off; off >>= 1) {
    if ((int)threadIdx.x < off) {
      sh[threadIdx.x] += sh[threadIdx.x + off];
      sh[256 + threadIdx.x] += sh[256 + threadIdx.x + off];
    }
    __syncthreads();
  }
  if (threadIdx.x == 0) {
    const double invB = 1.0 / (double)B;
    const double jtr_loss = sh[0] * invB / 42.0;            // mean over B*14*3
    const double loss = jtr_loss + 15.0 - sh[256] * invB;   // 15 - sum of cos means
    out[0] = (float)loss;
    out[1] = (float)jtr_loss;
  }
}

extern "C" void kernel_launch(void* const* d_in, const int* in_sizes, int n_in,
                              void* d_out, int out_size, void* d_ws, size_t ws_size,
                              hipStream_t stream) {
  (void)n_in; (void)out_size; (void)ws_size;
  const float* scale = (const float*)d_in[0];
  const int* si = (const int*)d_in[1];
  const int* di = (const int*)d_in[2];
  const float* pred = (const float*)d_in[3];
  const float* targ = (const float*)d_in[4];
  const long long B = (long long)in_sizes[3] / PRED_F;

  const long long nFull = B / TILE;
  int G = 1024;
  if (nFull < G) G = (int)(nFull > 0 ? nFull : 1);

  double* partial = (double*)d_ws;  // 2*G doubles, every slot written each call
  const size_t ldsBytes = (size_t)LDS_FLOATS * sizeof(float);

  orientation_loss_main<<<dim3(G), dim3(BLOCK), ldsBytes, stream>>>(
      scale, si, di, pred, targ, partial, B);
  orientation_loss_final<<<dim3(1), dim3(256), 0, stream>>>(
      partial, G, (float*)d_out, B);
}